// OrthonormalWaveletRegularization_47510928228981
// MI455X (gfx1250) — compile-verified
//
#include <hip/hip_runtime.h>

typedef __attribute__((ext_vector_type(2))) float v2f;
typedef __attribute__((ext_vector_type(8))) float v8f;

__device__ __forceinline__ float lane_bcast(float v, int src) {
  // gather across the wave32; lowers to ds_bpermute_b32 on gfx1250
  return __shfl(v, src, 32);
}

__device__ __forceinline__ float wave_sum32(float v) {
#pragma unroll
  for (int off = 16; off > 0; off >>= 1)
    v += __shfl_xor(v, off, 32);
  return v;
}

__global__ __launch_bounds__(32)
void OrthonormalWaveletRegularization_47510928228981_kernel(
    const float* __restrict__ h, const float* __restrict__ g,
    float* __restrict__ out) {
  const int lane = threadIdx.x;          // 0..31, one wave32
  const float hv = h[lane];
  const float gv = g[lane];

  const float SQRT2 = 1.41421356237309504880f;

  // ---- scalar terms: l1, l2, l3 via wave reductions ----
  const float sum_h  = wave_sum32(hv);
  const float sum_g  = wave_sum32(gv);
  const float dot_hh = wave_sum32(hv * hv);

  float l1 = sum_h - SQRT2; l1 *= l1;
  float l2 = sum_g * sum_g;
  float l3 = dot_hh - 1.0f; l3 *= l3;

  // Lane decomposition for f32 WMMA operands (16x4 A / 4x16 B layouts):
  // lanes 0-15 hold K = kb, kb+1 with kb = 4j; lanes 16-31 hold kb = 4j+2.
  const int half = (lane >= 16) ? 1 : 0;
  const int m    = lane & 15;            // matrix row index M carried by this lane (A side)

  // ---- l4: even-lag autocorrelation as 8x chained V_WMMA_F32_16X16X4_F32 ----
  // A[m,k] = h[k + 2*(m+1)] (0 beyond L), B[k,n] = h[k] broadcast across n.
  // C[m,n] = autocorr(2*(m+1)); all 16 columns identical.
  v8f acc4 = {0.f, 0.f, 0.f, 0.f, 0.f, 0.f, 0.f, 0.f};
#pragma unroll
  for (int j = 0; j < 8; ++j) {
    const int kb = 4 * j + 2 * half;     // K carried in vgpr0; vgpr1 carries kb+1
    const int shift = 2 * (m + 1);       // even lag for row m: 2..32
    int ia0 = kb + shift;
    int ia1 = kb + 1 + shift;
    float a0 = lane_bcast(hv, ia0 & 31); if (ia0 > 31) a0 = 0.0f;
    float a1 = lane_bcast(hv, ia1 & 31); if (ia1 > 31) a1 = 0.0f;
    float b0 = lane_bcast(hv, kb);       // B row kb, same for every column n
    float b1 = lane_bcast(hv, kb + 1);
    v2f a = {a0, a1};
    v2f b = {b0, b1};
    acc4 = __builtin_amdgcn_wmma_f32_16x16x4_f32(
        /*neg_a=*/false, a, /*neg_b=*/false, b,
        /*c_mod=*/(short)0, acc4, /*reuse_a=*/false, /*reuse_b=*/false);
  }
  float part4 = 0.0f;
#pragma unroll
  for (int v = 0; v < 8; ++v) part4 += acc4[v] * acc4[v];
  // Sum over full C tile counts each row 16x (identical columns); row M=15 is
  // lag 32 which is exactly zero (shift past end), matching k=1..15.
  const float l4 = wave_sum32(part4) * (1.0f / 16.0f);

  // ---- l5: Vandermonde moment matvec as 8x chained V_WMMA_F32_16X16X4_F32 ----
  // A[m,k] = k^m  (p = m+1, exponent p-1 = m);  B[k,n] = g[k] broadcast.
  v8f acc5 = {0.f, 0.f, 0.f, 0.f, 0.f, 0.f, 0.f, 0.f};
#pragma unroll
  for (int j = 0; j < 8; ++j) {
    const int kb = 4 * j + 2 * half;
    const float base0 = (float)kb;
    const float base1 = (float)(kb + 1);
    float a0 = 1.0f, a1 = 1.0f;
    for (int t = 0; t < m; ++t) { a0 *= base0; a1 *= base1; }  // k^m, 0^0 = 1
    float b0 = lane_bcast(gv, kb);
    float b1 = lane_bcast(gv, kb + 1);
    v2f a = {a0, a1};
    v2f b = {b0, b1};
    acc5 = __builtin_amdgcn_wmma_f32_16x16x4_f32(
        false, a, false, b, (short)0, acc5, false, false);
  }
  // Row M carried by acc5[v]: M = v (lanes 0-15) or v+8 (lanes 16-31); p = M+1.
  float part5 = 0.0f;
#pragma unroll
  for (int v = 0; v < 8; ++v) {
    const int M = v + 8 * half;
    const float norm = SQRT2 / (float)(1 << (M + 1));  // sqrt2 / 2^p
    const float t = acc5[v] * norm;
    part5 += t * t;
  }
  const float l5 = wave_sum32(part5) * (1.0f / 16.0f);

  if (lane == 0) out[0] = (l1 + l2) + (l3 + l4) + l5;
}

extern "C" void kernel_launch(void* const* d_in, const int* in_sizes, int n_in,
                              void* d_out, int out_size, void* d_ws, size_t ws_size,
                              hipStream_t stream) {
  (void)in_sizes; (void)n_in; (void)out_size; (void)d_ws; (void)ws_size;
  const float* h = (const float*)d_in[0];
  const float* g = (const float*)d_in[1];
  float* out = (float*)d_out;
  OrthonormalWaveletRegularization_47510928228981_kernel<<<1, 32, 0, stream>>>(h, g, out);
}